// LightGCN_89850715832674
// MI455X (gfx1250) — compile-verified
//
#include <hip/hip_runtime.h>
#include <cstdint>

// ---------------- problem constants (match reference) ----------------
#define N_USERS   200000
#define N_ITEMS   100000
#define N_NODESK  (N_USERS + N_ITEMS)     // 300000
#define EMB       64
#define N_EDGES   4000000
#define N_HOPS    3
#define SLICES    (N_HOPS + 1)            // 4
#define NODE_STRIDE (SLICES * EMB)        // 256 floats per node in d_out
#define EPB       1024                    // edges staged per block
#define PF_DIST   8                       // gather prefetch lookahead (edges)

// ---------------- small deterministic hash RNG ----------------
__device__ __forceinline__ uint32_t mix32(uint32_t x) {
    x ^= x >> 16; x *= 0x7feb352du;
    x ^= x >> 15; x *= 0x846ca68bu;
    x ^= x >> 16;
    return x;
}

// ---------------- CDNA5 async global->LDS DMA (ASYNCcnt path) ----------------
__device__ __forceinline__ void async_ld16(uint32_t lds_off, const void* gptr) {
    // per-lane: copies 16 bytes from the global address in gptr to the LDS address in lds_off, tracked on ASYNCcnt
    asm volatile("global_load_async_to_lds_b128 %0, %1, off"
                 :: "v"(lds_off), "v"((unsigned long long)(uintptr_t)gptr)
                 : "memory");
}
__device__ __forceinline__ void wait_async0() {
    asm volatile("s_wait_asynccnt 0" ::: "memory");
}

// ---------------- kernel: edge-dropout + SpMM scatter (one hop) ----------------
// Reads slice `hop`, atomically accumulates into slice `hop+1`.
// Placed FIRST in the TU so the disasm snippet shows the async/atomic codegen.
__global__ __launch_bounds__(256)
void lgcn_spmm(const int*   __restrict__ rows,
               const int*   __restrict__ cols,
               const float* __restrict__ vals,
               float*       __restrict__ out, int hop) {
    __shared__ int   s_row[EPB];
    __shared__ int   s_col[EPB];
    __shared__ float s_val[EPB];

    const uint32_t tid  = threadIdx.x;            // 0..255 = 8 wave32s
    const uint32_t base = blockIdx.x * EPB;

    // --- stage this block's edge tile into LDS with async DMA (16B per lane) ---
    uint32_t off = base + tid * 4u;               // each lane copies 4 elements
    if (off + 4u > (uint32_t)N_EDGES) off = N_EDGES - 4u;  // clamp tail (unused)
    async_ld16((uint32_t)(uintptr_t)&s_row[tid * 4u], rows + off);
    async_ld16((uint32_t)(uintptr_t)&s_col[tid * 4u], cols + off);
    async_ld16((uint32_t)(uintptr_t)&s_val[tid * 4u], vals + off);
    wait_async0();
    __syncthreads();

    const uint32_t wave = tid >> 5, lane = tid & 31u;
    // 32-bit element offsets: max index 300000*256 = 76.8M < 2^31
    const float* srcbase = out + hop * EMB + lane * 2u;        // + node*256
    float*       dstbase = out + (hop + 1) * EMB + lane * 2u;  // + node*256

    const uint32_t per_wave = EPB / 8u;           // 128 edges per wave
    const uint32_t i0 = wave * per_wave;
    const uint32_t hseed = 0x7F4A7C15u * (uint32_t)(hop + 1);

    for (uint32_t i = 0; i < per_wave; ++i) {
        uint32_t idx = i0 + i;
        uint32_t g   = base + idx;                // global edge id
        if (g >= (uint32_t)N_EDGES) break;

        // prefetch a future gather-source row (random access -> latency-bound);
        // emits global_prefetch_b8, no LOADcnt cost
        if (i + PF_DIST < per_wave) {
            uint32_t cn = (uint32_t)s_col[idx + PF_DIST];
            __builtin_prefetch(srcbase + cn * (uint32_t)NODE_STRIDE, 0, 0);
        }

        // edge dropout: keep with p = 0.5, scale kept vals by 1/(1-0.5) = 2
        uint32_t h = mix32(g * 0x9E3779B9u + hseed);
        if (!(h >> 31)) continue;

        float    v2 = s_val[idx] * 2.0f;          // uniform-addr LDS broadcast read
        uint32_t c  = (uint32_t)s_col[idx];
        uint32_t r  = (uint32_t)s_row[idx];

        float2 sv = *(const float2*)(srcbase + c * (uint32_t)NODE_STRIDE);
        float* d  = dstbase + r * (uint32_t)NODE_STRIDE;
        unsafeAtomicAdd(d + 0, sv.x * v2);        // global_atomic_add_f32, L2-resident
        unsafeAtomicAdd(d + 1, sv.y * v2);
    }
}

// ---------------- kernel: write hop-0 slice, zero slices 1..3 ----------------
__global__ __launch_bounds__(256)
void lgcn_init(const float* __restrict__ ue, const float* __restrict__ ie,
               float* __restrict__ out) {
    uint32_t idx = blockIdx.x * blockDim.x + threadIdx.x;  // [0, N*16) float4 units
    if (idx >= (uint32_t)N_NODESK * 16u) return;
    uint32_t n = idx >> 4, q = idx & 15;
    const float4* src = (n < N_USERS)
        ? (const float4*)ue + n * 16u + q
        : (const float4*)ie + (n - N_USERS) * 16u + q;
    float4* o = (float4*)out + n * (uint32_t)(NODE_STRIDE / 4) + q;
    o[0]  = *src;                         // embs[:,0,:] = all_embed
    float4 z = make_float4(0.f, 0.f, 0.f, 0.f);
    o[16] = z; o[32] = z; o[48] = z;      // zero accumulation slices for hops 1..3
}

// ---------------- kernel: message (dense) dropout on slice hop+1 ----------------
__global__ __launch_bounds__(256)
void lgcn_dropout(float* __restrict__ out, int hop) {
    uint32_t idx = blockIdx.x * blockDim.x + threadIdx.x;  // [0, N*16) float4 units
    if (idx >= (uint32_t)N_NODESK * 16u) return;
    uint32_t n = idx >> 4, q = idx & 15;
    float4* p = (float4*)(out + n * (uint32_t)NODE_STRIDE + (hop + 1) * EMB) + q;
    float4 v = *p;
    const float inv = 1.0f / 0.9f;
    const uint32_t e0 = idx * 4u;
    const uint32_t hs = 0x85EBCA6Bu * (uint32_t)(hop + 1);
    float* f = (float*)&v;
#pragma unroll
    for (int k = 0; k < 4; ++k) {
        uint32_t h = mix32((e0 + (uint32_t)k) ^ hs ^ 0xC2B2AE35u);
        f[k] = (h >= 429496730u) ? f[k] * inv : 0.0f;  // keep prob ~0.9, scale 1/0.9
    }
    *p = v;
}

// ---------------- host-side launcher ----------------
extern "C" void kernel_launch(void* const* d_in, const int* in_sizes, int n_in,
                              void* d_out, int out_size, void* d_ws, size_t ws_size,
                              hipStream_t stream) {
    (void)in_sizes; (void)n_in; (void)out_size; (void)d_ws; (void)ws_size;
    const float* ue   = (const float*)d_in[0];
    const float* ie   = (const float*)d_in[1];
    const int*   rows = (const int*)  d_in[2];
    const int*   cols = (const int*)  d_in[3];
    const float* vals = (const float*)d_in[4];
    float* out = (float*)d_out;

    const int elem4       = N_NODESK * 16;                 // float4 work-items
    const int elem_blocks = (elem4 + 255) / 256;
    const int spmm_blocks = (N_EDGES + EPB - 1) / EPB;

    lgcn_init<<<elem_blocks, 256, 0, stream>>>(ue, ie, out);
    for (int hop = 0; hop < N_HOPS; ++hop) {
        lgcn_spmm<<<spmm_blocks, 256, 0, stream>>>(rows, cols, vals, out, hop);
        lgcn_dropout<<<elem_blocks, 256, 0, stream>>>(out, hop);
    }
}